// Graph_Net_78761110274299
// MI455X (gfx1250) — compile-verified
//
#include <hip/hip_runtime.h>

#define N_NODES 50000
#define N_EDGES 800000
#define G_GRAPHS 128
#define IN_C 64
#define HID 128
#define HID2 256
#define OUT_C 16
#define EDGE_C 16
#define N_LAYERS 3
#define BN_EPS 1e-5f

typedef float v2f __attribute__((ext_vector_type(2)));
typedef float v8f __attribute__((ext_vector_type(8)));

__device__ __forceinline__ v8f wmma_f32_k4(v8f acc, v2f a, v2f b) {
    // D = A(16x4) x B(4x16) + C, fp32, wave32.
    // args: (neg_a, A, neg_b, B, c_mod, C, reuse_a, reuse_b)
    return __builtin_amdgcn_wmma_f32_16x16x4_f32(false, a, false, b, (short)0, acc,
                                                 false, false);
}

__device__ __forceinline__ void atomic_add_f32(float* p, float v) {
    __hip_atomic_fetch_add(p, v, __ATOMIC_RELAXED, __HIP_MEMORY_SCOPE_AGENT);
}

__device__ __forceinline__ v2f ldg2(const float* __restrict__ p) {
    return *(const v2f*)p;  // 8B-aligned contiguous pair -> global_load_b64
}

// ---------------------------------------------------------------------------
// Zero fill
// ---------------------------------------------------------------------------
__global__ void k_zero(float* __restrict__ p, int n) {
    int i = blockIdx.x * blockDim.x + threadIdx.x;
    if (i < n) p[i] = 0.0f;
}

// ---------------------------------------------------------------------------
// Transpose [R,C] -> [C,R] (weights are tiny & L2-resident; one-time cost)
// ---------------------------------------------------------------------------
__global__ void k_transpose(const float* __restrict__ in, float* __restrict__ out,
                            int R, int C) {
    int idx = blockIdx.x * blockDim.x + threadIdx.x;
    if (idx >= R * C) return;
    int r = idx / C, c = idx - r * C;
    out[(size_t)c * R + r] = in[idx];
}

// ---------------------------------------------------------------------------
// h = x @ W0 + b0    [N,64] x [64,128] -> [N,128]
// W0t is transposed [128][64] so each lane's K-pair is one b64 load.
// block: 256 thr = 8 waves; block owns 16 rows; wave w owns cols [16w,16w+16)
// ---------------------------------------------------------------------------
__global__ __launch_bounds__(256) void k_init_gemm(const float* __restrict__ x,
                                                   const float* __restrict__ W0t,
                                                   const float* __restrict__ b,
                                                   float* __restrict__ h) {
    __shared__ __align__(16) float sA[16][IN_C];  // 4 KB
    const int tid  = threadIdx.x;
    const int wave = tid >> 5, lane = tid & 31;
    const int m0   = blockIdx.x * 16;

    // cooperative A-tile load: 16*64 = 1024 floats = 256 float4
    {
        const float4* src4 = (const float4*)(x + (size_t)m0 * IN_C);
        ((float4*)&sA[0][0])[tid] = src4[tid];
    }
    __syncthreads();

    const int half = lane >> 4, col = lane & 15;
    const int n0 = wave * 16;
    const float* brow = W0t + (size_t)(n0 + col) * IN_C;  // column (n0+col), K-major
    v8f acc = {};
    for (int k = 0; k < IN_C; k += 4) {
        const int ka = k + 2 * half;
        v2f a; a.x = sA[col][ka]; a.y = sA[col][ka + 1];
        acc = wmma_f32_k4(acc, a, ldg2(brow + ka));
    }
    const int mb = half * 8;
    const float bias = b[n0 + col];
#pragma unroll
    for (int r = 0; r < 8; ++r)
        h[(size_t)(m0 + mb + r) * HID + n0 + col] = acc[r] + bias;
}

// ---------------------------------------------------------------------------
// Edge message pass:  agg[dst] += relu(h[src] + edge_attr @ We + be)
// one wave per edge; lane handles 4 channels; We (16x128) cached in LDS
// ---------------------------------------------------------------------------
__global__ __launch_bounds__(256) void k_edge(const float* __restrict__ h,
                                              const float* __restrict__ ea,
                                              const int* __restrict__ src,
                                              const int* __restrict__ dst,
                                              const float* __restrict__ We,
                                              const float* __restrict__ be,
                                              float* __restrict__ agg) {
    __shared__ float sW[EDGE_C * HID];  // 8 KB
    __shared__ float sb[HID];
    const int tid = threadIdx.x;
    for (int i = tid; i < EDGE_C * HID; i += 256) sW[i] = We[i];
    if (tid < HID) sb[tid] = be[tid];
    __syncthreads();

    const int wave = tid >> 5, lane = tid & 31;
    const int e = blockIdx.x * 8 + wave;
    if (e >= N_EDGES) return;

    const int s = src[e], d = dst[e];
    const int c0 = lane * 4;
    const float4 hv = *(const float4*)(h + (size_t)s * HID + c0);

    float a0 = sb[c0], a1 = sb[c0 + 1], a2 = sb[c0 + 2], a3 = sb[c0 + 3];
    const float4* earow = (const float4*)(ea + (size_t)e * EDGE_C);
#pragma unroll
    for (int kk = 0; kk < 4; ++kk) {
        const float4 av = earow[kk];
        const float ak[4] = {av.x, av.y, av.z, av.w};
#pragma unroll
        for (int j = 0; j < 4; ++j) {
            const float* wr = &sW[(kk * 4 + j) * HID + c0];
            a0 = fmaf(ak[j], wr[0], a0);
            a1 = fmaf(ak[j], wr[1], a1);
            a2 = fmaf(ak[j], wr[2], a2);
            a3 = fmaf(ak[j], wr[3], a3);
        }
    }
    float* ap = agg + (size_t)d * HID + c0;
    atomic_add_f32(ap + 0, fmaxf(hv.x + a0, 0.0f));
    atomic_add_f32(ap + 1, fmaxf(hv.y + a1, 0.0f));
    atomic_add_f32(ap + 2, fmaxf(hv.z + a2, 0.0f));
    atomic_add_f32(ap + 3, fmaxf(hv.w + a3, 0.0f));
}

// ---------------------------------------------------------------------------
// Fused GIN MLP:  z = (1+eps)h + agg ; z2 = relu(z@W1+b1)@W2 + b2
// W1t [256][128], W2t [128][256] are K-major so B pairs load as b64.
// block = 256 thr (8 waves), owns 16 rows.  z2 written over agg (safe: rows
// are read into LDS before the write, and blocks own disjoint row ranges).
// ---------------------------------------------------------------------------
__global__ __launch_bounds__(256) void k_mlp(const float* __restrict__ h,
                                             const float* __restrict__ agg,
                                             const float* __restrict__ eps_i,
                                             const float* __restrict__ W1t,
                                             const float* __restrict__ b1,
                                             const float* __restrict__ W2t,
                                             const float* __restrict__ b2,
                                             float* __restrict__ z2out) {
    __shared__ __align__(16) float sZ[16][HID];   // 8 KB
    __shared__ __align__(16) float sT[16][HID2];  // 16 KB
    const int tid  = threadIdx.x;
    const int wave = tid >> 5, lane = tid & 31;
    const int m0   = blockIdx.x * 16;
    const float ep = 1.0f + eps_i[0];

    // stage z = (1+eps)*h + agg : 16*128 = 512 float4
    {
        const float4* h4 = (const float4*)(h + (size_t)m0 * HID);
        const float4* a4 = (const float4*)(agg + (size_t)m0 * HID);
        float4* z4 = (float4*)&sZ[0][0];
        for (int i = tid; i < 512; i += 256) {
            const float4 hv = h4[i], av = a4[i];
            float4 z;
            z.x = fmaf(ep, hv.x, av.x);
            z.y = fmaf(ep, hv.y, av.y);
            z.z = fmaf(ep, hv.z, av.z);
            z.w = fmaf(ep, hv.w, av.w);
            z4[i] = z;
        }
    }
    __syncthreads();

    const int half = lane >> 4, col = lane & 15, mb = half * 8;

    // phase 1: t = relu(z@W1+b1)  [16,128]x[128,256]; wave covers 2 col-tiles
    for (int t = wave; t < 16; t += 8) {
        const int n0 = t * 16;
        const float* brow = W1t + (size_t)(n0 + col) * HID;
        v8f acc = {};
        for (int k = 0; k < HID; k += 4) {
            const int ka = k + 2 * half;
            v2f a; a.x = sZ[col][ka]; a.y = sZ[col][ka + 1];
            acc = wmma_f32_k4(acc, a, ldg2(brow + ka));
        }
        const float bias = b1[n0 + col];
#pragma unroll
        for (int r = 0; r < 8; ++r)
            sT[mb + r][n0 + col] = fmaxf(acc[r] + bias, 0.0f);
    }
    __syncthreads();

    // phase 2: z2 = t@W2+b2  [16,256]x[256,128]; wave covers 1 col-tile
    {
        const int n0 = wave * 16;
        const float* brow = W2t + (size_t)(n0 + col) * HID2;
        v8f acc = {};
        for (int k = 0; k < HID2; k += 4) {
            const int ka = k + 2 * half;
            v2f a; a.x = sT[col][ka]; a.y = sT[col][ka + 1];
            acc = wmma_f32_k4(acc, a, ldg2(brow + ka));
        }
        const float bias = b2[n0 + col];
#pragma unroll
        for (int r = 0; r < 8; ++r)
            z2out[(size_t)(m0 + mb + r) * HID + n0 + col] = acc[r] + bias;
    }
}

// ---------------------------------------------------------------------------
// BN stats: per-channel sum & sumsq over all nodes -> stats[0:128], stats[128:256]
// ---------------------------------------------------------------------------
__global__ __launch_bounds__(256) void k_bn_stats(const float* __restrict__ z,
                                                  float* __restrict__ stats) {
    __shared__ float ssum[HID], ssq[HID];
    const int tid = threadIdx.x;
    const int c = tid & (HID - 1), half = tid >> 7;
    float s = 0.0f, q = 0.0f;
    for (int r = blockIdx.x * 2 + half; r < N_NODES; r += gridDim.x * 2) {
        const float v = z[(size_t)r * HID + c];
        s += v;
        q = fmaf(v, v, q);
    }
    if (half) { ssum[c] = s; ssq[c] = q; }
    __syncthreads();
    if (!half) {
        atomic_add_f32(&stats[c], s + ssum[c]);
        atomic_add_f32(&stats[HID + c], q + ssq[c]);
    }
}

// ---------------------------------------------------------------------------
// BN apply + ReLU:  h = relu(gamma*(z-mu)*rsqrt(var+eps)+beta)
// ---------------------------------------------------------------------------
__global__ __launch_bounds__(256) void k_bn_apply(const float* __restrict__ z,
                                                  const float* __restrict__ stats,
                                                  const float* __restrict__ gamma,
                                                  const float* __restrict__ beta,
                                                  float* __restrict__ hout) {
    const int idx = blockIdx.x * 256 + threadIdx.x;
    if (idx >= N_NODES * HID) return;
    const int c = idx & (HID - 1);
    const float invn = 1.0f / (float)N_NODES;
    const float mu  = stats[c] * invn;
    const float var = stats[HID + c] * invn - mu * mu;
    const float inv = rsqrtf(var + BN_EPS);
    hout[idx] = fmaxf(gamma[c] * (z[idx] - mu) * inv + beta[c], 0.0f);
}

// ---------------------------------------------------------------------------
// Global mean pool (sum + counts via atomics)
// ---------------------------------------------------------------------------
__global__ __launch_bounds__(256) void k_pool(const float* __restrict__ h,
                                              const int* __restrict__ batch,
                                              float* __restrict__ pooled,
                                              float* __restrict__ counts) {
    const int idx = blockIdx.x * 256 + threadIdx.x;
    if (idx >= N_NODES * HID) return;
    const int c = idx & (HID - 1);
    const int node = idx >> 7;
    const int g = batch[node];
    atomic_add_f32(&pooled[(size_t)g * HID + c], h[idx]);
    if (c == 0) atomic_add_f32(&counts[g], 1.0f);
}

// ---------------------------------------------------------------------------
// out = (pooled/count) @ Wf + bf     [128,128]x[128,16]
// ---------------------------------------------------------------------------
__global__ __launch_bounds__(256) void k_final(const float* __restrict__ pooled,
                                               const float* __restrict__ counts,
                                               const float* __restrict__ Wf,
                                               const float* __restrict__ bf,
                                               float* __restrict__ out) {
    const int idx = blockIdx.x * 256 + threadIdx.x;
    if (idx >= G_GRAPHS * OUT_C) return;
    const int g = idx >> 4, o = idx & 15;
    const float cnt = fmaxf(counts[g], 1.0f);
    float s = 0.0f;
    for (int c = 0; c < HID; ++c) s = fmaf(pooled[(size_t)g * HID + c], Wf[c * OUT_C + o], s);
    out[idx] = s / cnt + bf[o];
}

// ---------------------------------------------------------------------------
extern "C" void kernel_launch(void* const* d_in, const int* in_sizes, int n_in,
                              void* d_out, int out_size, void* d_ws, size_t ws_size,
                              hipStream_t stream) {
    const float* x          = (const float*)d_in[0];
    const int*   edge_index = (const int*)d_in[1];
    const int*   batch      = (const int*)d_in[2];
    const float* edge_attr  = (const float*)d_in[3];
    const float* lin_init_W = (const float*)d_in[4];
    const float* lin_init_b = (const float*)d_in[5];
    const float* edge_W     = (const float*)d_in[6];
    const float* edge_b     = (const float*)d_in[7];
    const float* mlp_W1     = (const float*)d_in[8];
    const float* mlp_b1     = (const float*)d_in[9];
    const float* mlp_W2     = (const float*)d_in[10];
    const float* mlp_b2     = (const float*)d_in[11];
    const float* eps        = (const float*)d_in[12];
    const float* bn_gamma   = (const float*)d_in[13];
    const float* bn_beta    = (const float*)d_in[14];
    const float* lin_fW     = (const float*)d_in[15];
    const float* lin_fb     = (const float*)d_in[16];
    float* out = (float*)d_out;

    // workspace layout
    float* buf_h   = (float*)d_ws;                       // N*128
    float* buf_agg = buf_h + (size_t)N_NODES * HID;      // N*128 (also z2)
    float* stats   = buf_agg + (size_t)N_NODES * HID;    // 256
    float* pooled  = stats + 2 * HID;                    // G*128
    float* counts  = pooled + (size_t)G_GRAPHS * HID;    // G
    float* w0t     = counts + G_GRAPHS;                  // 128*64
    float* w1t     = w0t + (size_t)HID * IN_C;           // L*256*128 (K-major)
    float* w2t     = w1t + (size_t)N_LAYERS * HID2 * HID;// L*128*256 (K-major)

    const int src_off = 0, dst_off = N_EDGES;
    const int nh = N_NODES * HID;
    const int nh_blocks = (nh + 255) / 256;
    const int wt_blocks = (HID * HID2 + 255) / 256;

    // one-time (per launch) weight transposes into K-major layout
    k_transpose<<<(IN_C * HID + 255) / 256, 256, 0, stream>>>(lin_init_W, w0t, IN_C, HID);
    for (int i = 0; i < N_LAYERS; ++i) {
        k_transpose<<<wt_blocks, 256, 0, stream>>>(mlp_W1 + (size_t)i * HID * HID2,
                                                   w1t + (size_t)i * HID2 * HID, HID, HID2);
        k_transpose<<<wt_blocks, 256, 0, stream>>>(mlp_W2 + (size_t)i * HID2 * HID,
                                                   w2t + (size_t)i * HID * HID2, HID2, HID);
    }

    k_init_gemm<<<N_NODES / 16, 256, 0, stream>>>(x, w0t, lin_init_b, buf_h);

    for (int i = 0; i < N_LAYERS; ++i) {
        k_zero<<<nh_blocks, 256, 0, stream>>>(buf_agg, nh);
        k_edge<<<N_EDGES / 8, 256, 0, stream>>>(buf_h, edge_attr,
                                                edge_index + src_off, edge_index + dst_off,
                                                edge_W + (size_t)i * EDGE_C * HID,
                                                edge_b + (size_t)i * HID, buf_agg);
        k_mlp<<<N_NODES / 16, 256, 0, stream>>>(buf_h, buf_agg, eps + i,
                                                w1t + (size_t)i * HID2 * HID,
                                                mlp_b1 + (size_t)i * HID2,
                                                w2t + (size_t)i * HID * HID2,
                                                mlp_b2 + (size_t)i * HID, buf_agg);
        k_zero<<<1, 256, 0, stream>>>(stats, 2 * HID);
        k_bn_stats<<<512, 256, 0, stream>>>(buf_agg, stats);
        k_bn_apply<<<nh_blocks, 256, 0, stream>>>(buf_agg, stats,
                                                  bn_gamma + (size_t)i * HID,
                                                  bn_beta + (size_t)i * HID, buf_h);
    }

    k_zero<<<(G_GRAPHS * HID + G_GRAPHS + 255) / 256, 256, 0, stream>>>(
        pooled, G_GRAPHS * HID + G_GRAPHS);
    k_pool<<<nh_blocks, 256, 0, stream>>>(buf_h, batch, pooled, counts);
    k_final<<<(G_GRAPHS * OUT_C + 255) / 256, 256, 0, stream>>>(pooled, counts,
                                                                lin_fW, lin_fb, out);
}